// a_max_op_6631429505489
// MI455X (gfx1250) — compile-verified
//
#include <hip/hip_runtime.h>
#include <hip/hip_bf16.h>
#include <stdint.h>

typedef float v2f __attribute__((ext_vector_type(2)));
typedef float v8f __attribute__((ext_vector_type(8)));

#define DIM 128
#define KSTEPS 32            // 128 / 4 (K per V_WMMA_F32_16X16X4_F32)
#define NT 2                 // 16-col tiles per wave
#define COLS_PER_WAVE 32     // NT * 16
#define BLOCK_THREADS 128    // 4 waves cover all 128 columns

// Fill output with -inf bit pattern (identity of segment_max for floats).
__global__ __launch_bounds__(256) void fill_neginf_kernel(int* out, int n) {
    for (int i = blockIdx.x * blockDim.x + threadIdx.x; i < n;
         i += gridDim.x * blockDim.x) {
        out[i] = 0xFF800000;  // -inf
    }
}

// msg = relu(src @ W^T + b); out[dst[e], :] = max over edges (via i32 atomicMax).
__global__ __launch_bounds__(BLOCK_THREADS, 1) void edge_gemm_segmax_kernel(
    const float* __restrict__ src,   // [E, 128]
    const float* __restrict__ W,     // [128, 128] row-major [f][d]
    const float* __restrict__ bias,  // [128]
    const int*   __restrict__ dst,   // [E]
    float* __restrict__ out,         // [N, 128], pre-filled with -inf
    int n_tiles)                     // E / 16
{
    const int lane = threadIdx.x & 31;
    const int wave = threadIdx.x >> 5;
    const int half = lane >> 4;   // 0: lanes 0-15, 1: lanes 16-31
    const int l16  = lane & 15;

    // ---- Preload B = W^T fragments for this wave's 32 columns, all K. ----
    // B frag (4x16) for (nt, ks): lane holds B[k, n] = W[n, k] with
    // n = wave*32 + nt*16 + l16, k = ks*4 + half*2 + {0,1} (contiguous in W row).
    v2f bfrag[NT][KSTEPS];
    float bias_c[NT];
#pragma unroll
    for (int nt = 0; nt < NT; ++nt) {
        const int col = wave * COLS_PER_WAVE + nt * 16 + l16;
        const float* wrow = W + (size_t)col * DIM + half * 2;
#pragma unroll
        for (int ks = 0; ks < KSTEPS; ++ks)
            bfrag[nt][ks] = *(const v2f*)(wrow + ks * 4);
        bias_c[nt] = bias[col];
    }

    // ---- Grid-stride over 16-edge tiles. ----
    for (int t = blockIdx.x; t < n_tiles; t += gridDim.x) {
        // A frag (16x4) for ks: lane holds A[m, k], m = t*16 + l16,
        // k = ks*4 + half*2 + {0,1}.
        const float* arow = src + (size_t)(t * 16 + l16) * DIM + half * 2;
        v2f afrag[KSTEPS];
#pragma unroll
        for (int ks = 0; ks < KSTEPS; ++ks)
            afrag[ks] = *(const v2f*)(arow + ks * 4);

        v8f acc[NT] = {};
#pragma unroll
        for (int ks = 0; ks < KSTEPS; ++ks) {
#pragma unroll
            for (int nt = 0; nt < NT; ++nt) {
                // 8 args: (neg_a, A, neg_b, B, c_mod, C, reuse_a, reuse_b)
                acc[nt] = __builtin_amdgcn_wmma_f32_16x16x4_f32(
                    false, afrag[ks], false, bfrag[nt][ks],
                    (short)0, acc[nt], false, false);
            }
        }

        // ---- Epilogue: bias + relu, then integer atomic-max scatter. ----
        // C layout: VGPR r holds (M = r + 8*half, N = l16).
#pragma unroll
        for (int r = 0; r < 8; ++r) {
            const int e    = t * 16 + r + half * 8;
            const int node = dst[e];
            int* orow = (int*)out + (size_t)node * DIM
                        + wave * COLS_PER_WAVE + l16;
#pragma unroll
            for (int nt = 0; nt < NT; ++nt) {
                float v = acc[nt][r] + bias_c[nt];
                v = v > 0.0f ? v : 0.0f;
                // Non-negative floats order identically to their i32 bit
                // patterns; -inf (0xFF800000) is a very negative i32.
                atomicMax(orow + nt * 16, __float_as_int(v));
            }
        }
    }
}

extern "C" void kernel_launch(void* const* d_in, const int* in_sizes, int n_in,
                              void* d_out, int out_size, void* d_ws, size_t ws_size,
                              hipStream_t stream) {
    const float* src  = (const float*)d_in[0];   // [E,128]
    const float* W    = (const float*)d_in[1];   // [128,128]
    const float* bias = (const float*)d_in[2];   // [128]
    const int*   dst  = (const int*)d_in[3];     // [E]
    float* out = (float*)d_out;                  // [N,128]

    const int E = in_sizes[0] / DIM;
    const int n_tiles = E / 16;                  // E is a multiple of 16

    // 1) out = -inf everywhere (segment_max identity).
    {
        int n = out_size;
        int blocks = (n + 255) / 256;
        if (blocks > 4096) blocks = 4096;
        fill_neginf_kernel<<<blocks, 256, 0, stream>>>((int*)out, n);
    }

    // 2) Fused GEMM + relu + segment-max scatter.
    {
        int blocks = n_tiles < 4096 ? n_tiles : 4096;
        edge_gemm_segmax_kernel<<<blocks, BLOCK_THREADS, 0, stream>>>(
            src, W, bias, dst, out, n_tiles);
    }
}